// ClinicalGCN_67757404062361
// MI455X (gfx1250) — compile-verified
//
#include <hip/hip_runtime.h>
#include <hip/hip_bf16.h>
#include <cstddef>

#define N_NODES 50000
#define N_EDGES 800000
#define F_IN    256
#define F_H     128
#define F_C     4
#define ROW_TILES 3125   // 50000 / 16, exact

typedef float v2f __attribute__((ext_vector_type(2)));
typedef float v8f __attribute__((ext_vector_type(8)));

// ---------------- degree / normalization ----------------

__global__ void k_init_deg(float* __restrict__ deg) {
    int i = blockIdx.x * blockDim.x + threadIdx.x;
    if (i < N_NODES) deg[i] = 1.0f;   // self-loop contribution
}

__global__ void k_accum_deg(const int* __restrict__ dst, float* __restrict__ deg) {
    int e = blockIdx.x * blockDim.x + threadIdx.x;
    if (e < N_EDGES) atomicAdd(&deg[dst[e]], 1.0f);
}

__global__ void k_rsqrt(float* __restrict__ deg) {
    int i = blockIdx.x * blockDim.x + threadIdx.x;
    if (i < N_NODES) deg[i] = rsqrtf(deg[i]);   // deg >= 1 always
}

// ---------------- GEMM1: XW = x(50000x256) @ W1(256x128), fp32 WMMA ----------------
// One wave per 16-row tile; each wave accumulates all 8 col-tiles of H=128.

__global__ void __launch_bounds__(256)
k_gemm1(const float* __restrict__ x, const float* __restrict__ w1,
        float* __restrict__ xw) {
    const int lane = threadIdx.x & 31;
    const int wave = threadIdx.x >> 5;
    const int tile = blockIdx.x * 8 + wave;
    if (tile >= ROW_TILES) return;           // uniform per wave -> EXEC all-ones inside
    const int m0   = tile * 16;
    const int half = lane >> 4;              // 0: K even pair, 1: K odd pair
    const int col  = lane & 15;
    const int row  = m0 + col;               // A rows: lanes 0-15 and 16-31 both map M=0..15

    v8f acc[8];
#pragma unroll
    for (int n = 0; n < 8; ++n) acc[n] = (v8f)0.0f;

    const float* arow = x + (size_t)row * F_IN + 2 * half;

    for (int k0 = 0; k0 < F_IN; k0 += 4) {
        v2f a;
        a[0] = arow[k0];                     // K = k0 + 2*half
        a[1] = arow[k0 + 1];                 // K = k0 + 2*half + 1
#pragma unroll
        for (int n = 0; n < 8; ++n) {
            const float* bp = w1 + (size_t)(k0 + 2 * half) * F_H + n * 16 + col;
            v2f b;
            b[0] = bp[0];                    // B row K = k0+2*half, col
            b[1] = bp[F_H];                  // B row K = k0+2*half+1, col
            acc[n] = __builtin_amdgcn_wmma_f32_16x16x4_f32(
                false, a, false, b, (short)0, acc[n], false, false);
        }
    }

#pragma unroll
    for (int n = 0; n < 8; ++n) {
#pragma unroll
        for (int r = 0; r < 8; ++r) {
            int rr = m0 + half * 8 + r;      // C/D layout: VGPR r -> M = r + 8*half
            xw[(size_t)rr * F_H + n * 16 + col] = acc[n][r];
        }
    }
}

// ---------------- agg init with self-loop term (also zero-fills) ----------------

__global__ void k_init_agg(const float4* __restrict__ xw4, const float* __restrict__ dinv,
                           float4* __restrict__ agg4) {
    int idx = blockIdx.x * blockDim.x + threadIdx.x;   // N*32 float4 chunks, exact grid
    int i = idx >> 5;
    float w = dinv[i]; w *= w;
    float4 v = xw4[idx];
    float4 r; r.x = v.x * w; r.y = v.y * w; r.z = v.z * w; r.w = v.w * w;
    agg4[idx] = r;
}

// ---------------- edge scatter layer 1: 32 lanes x float4 per edge ----------------

__global__ void k_scatter1(const float* __restrict__ xw, const float* __restrict__ dinv,
                           const int* __restrict__ src, const int* __restrict__ dst,
                           float* __restrict__ agg) {
    unsigned idx = blockIdx.x * blockDim.x + threadIdx.x;  // E*32 threads, exact grid
    unsigned e = idx >> 5;
    unsigned c = idx & 31;
    if (e >= N_EDGES) return;
    int s = src[e], d = dst[e];
    float w = dinv[s] * dinv[d];
    const float4 v = *(const float4*)(xw + (size_t)s * F_H + c * 4);
    float* p = agg + (size_t)d * F_H + c * 4;
    atomicAdd(p + 0, v.x * w);
    atomicAdd(p + 1, v.y * w);
    atomicAdd(p + 2, v.z * w);
    atomicAdd(p + 3, v.w * w);
}

// ---------------- h = relu(agg + b1), in place ----------------

__global__ void k_relu_bias(float* __restrict__ h, const float* __restrict__ b1) {
    int idx = blockIdx.x * blockDim.x + threadIdx.x;   // N*H, exact grid
    float v = h[idx] + b1[idx & (F_H - 1)];
    h[idx] = v > 0.0f ? v : 0.0f;
}

// ---------------- GEMM2: H2 = h(50000x128) @ W2(128x4), fp32 WMMA, B padded to 16 ----

__global__ void __launch_bounds__(256)
k_gemm2(const float* __restrict__ h, const float* __restrict__ w2,
        float* __restrict__ h2) {
    const int lane = threadIdx.x & 31;
    const int wave = threadIdx.x >> 5;
    const int tile = blockIdx.x * 8 + wave;
    if (tile >= ROW_TILES) return;
    const int m0   = tile * 16;
    const int half = lane >> 4;
    const int col  = lane & 15;
    const int row  = m0 + col;

    v8f acc = (v8f)0.0f;
    const float* arow = h + (size_t)row * F_H + 2 * half;

    for (int k0 = 0; k0 < F_H; k0 += 4) {
        v2f a;
        a[0] = arow[k0];
        a[1] = arow[k0 + 1];
        v2f b;
        if (col < F_C) {
            const float* bp = w2 + (size_t)(k0 + 2 * half) * F_C + col;
            b[0] = bp[0];
            b[1] = bp[F_C];
        } else {
            b[0] = 0.0f;
            b[1] = 0.0f;
        }
        acc = __builtin_amdgcn_wmma_f32_16x16x4_f32(
            false, a, false, b, (short)0, acc, false, false);
    }

    if (col < F_C) {
#pragma unroll
        for (int r = 0; r < 8; ++r) {
            int rr = m0 + half * 8 + r;
            h2[(size_t)rr * F_C + col] = acc[r];
        }
    }
}

// ---------------- out init: self-loop term + bias ----------------

__global__ void k_init_out(const float* __restrict__ h2, const float* __restrict__ dinv,
                           const float* __restrict__ b2, float* __restrict__ out) {
    int idx = blockIdx.x * blockDim.x + threadIdx.x;
    if (idx < N_NODES * F_C) {
        int i = idx >> 2;
        float w = dinv[i]; w *= w;
        out[idx] = h2[idx] * w + b2[idx & 3];
    }
}

// ---------------- edge scatter layer 2: thread per edge, float4 ----------------

__global__ void k_scatter2(const float4* __restrict__ h2, const float* __restrict__ dinv,
                           const int* __restrict__ src, const int* __restrict__ dst,
                           float* __restrict__ out) {
    int e = blockIdx.x * blockDim.x + threadIdx.x;   // E threads, exact grid
    if (e < N_EDGES) {
        int s = src[e], d = dst[e];
        float w = dinv[s] * dinv[d];
        float4 v = h2[s];
        float* p = out + (size_t)d * F_C;
        atomicAdd(p + 0, v.x * w);
        atomicAdd(p + 1, v.y * w);
        atomicAdd(p + 2, v.z * w);
        atomicAdd(p + 3, v.w * w);
    }
}

extern "C" void kernel_launch(void* const* d_in, const int* in_sizes, int n_in,
                              void* d_out, int out_size, void* d_ws, size_t ws_size,
                              hipStream_t stream) {
    (void)in_sizes; (void)n_in; (void)out_size; (void)ws_size;

    const float* x  = (const float*)d_in[0];
    const int*   ei = (const int*)d_in[1];       // [2,E] row-major: src then dst
    const float* W1 = (const float*)d_in[2];
    const float* b1 = (const float*)d_in[3];
    const float* W2 = (const float*)d_in[4];
    const float* b2 = (const float*)d_in[5];
    float* out = (float*)d_out;

    const int* src = ei;
    const int* dst = ei + N_EDGES;

    // workspace layout (floats): deg/dinv | bufA (XW, later H2) | bufB (agg -> h)
    float* deg  = (float*)d_ws;
    float* bufA = deg + 50176;                        // 16B-aligned past N=50000
    float* bufB = bufA + (size_t)N_NODES * F_H;

    k_init_deg <<<(N_NODES + 255) / 256, 256, 0, stream>>>(deg);
    k_accum_deg<<<N_EDGES / 256,         256, 0, stream>>>(dst, deg);
    k_rsqrt    <<<(N_NODES + 255) / 256, 256, 0, stream>>>(deg);

    k_gemm1    <<<(ROW_TILES + 7) / 8,   256, 0, stream>>>(x, W1, bufA);

    k_init_agg <<<N_NODES * 32 / 256,    256, 0, stream>>>((const float4*)bufA, deg,
                                                           (float4*)bufB);
    k_scatter1 <<<N_EDGES * 32 / 256,    256, 0, stream>>>(bufA, deg, src, dst, bufB);
    k_relu_bias<<<N_NODES * F_H / 256,   256, 0, stream>>>(bufB, b1);

    k_gemm2    <<<(ROW_TILES + 7) / 8,   256, 0, stream>>>(bufB, W2, bufA);

    k_init_out <<<(N_NODES * F_C + 255) / 256, 256, 0, stream>>>(bufA, deg, b2, out);
    k_scatter2 <<<N_EDGES / 256,         256, 0, stream>>>((const float4*)bufA, deg,
                                                           src, dst, out);
}